// DGCNN_Embedder_11020886081804
// MI455X (gfx1250) — compile-verified
//
#include <hip/hip_runtime.h>

typedef float v2f __attribute__((ext_vector_type(2)));
typedef float v8f __attribute__((ext_vector_type(8)));
typedef int   vi4 __attribute__((vector_size(16)));

#define B_   4
#define N_   4096
#define KK_  20
#define NEG_SLOPE_ 0.2f
#define EPS_ 1e-5f

#if __has_builtin(__builtin_amdgcn_global_load_async_to_lds_b32) && \
    __has_builtin(__builtin_amdgcn_global_load_async_to_lds_b128) && \
    __has_builtin(__builtin_amdgcn_s_wait_asynccnt)
#define HAVE_ASYNC_LDS 1
#else
#define HAVE_ASYNC_LDS 0
#endif

static __device__ __forceinline__ v8f wmma_f32x4(v2f a, v2f b, v8f c) {
    return __builtin_amdgcn_wmma_f32_16x16x4_f32(false, a, false, b, (short)0, c, false, false);
}

// async global->LDS copies (ASYNCcnt path); safe fallback when builtin absent
static __device__ __forceinline__ void async_b32(const float* g, float* l) {
#if HAVE_ASYNC_LDS
    typedef __attribute__((address_space(1))) int gi_t;
    typedef __attribute__((address_space(3))) int li_t;
    __builtin_amdgcn_global_load_async_to_lds_b32((gi_t*)g, (li_t*)l, 0, 0);
#else
    *l = *g;
#endif
}
static __device__ __forceinline__ void async_b128(const float* g, float* l) {
#if HAVE_ASYNC_LDS
    typedef __attribute__((address_space(1))) vi4 gv_t;
    typedef __attribute__((address_space(3))) vi4 lv_t;
    __builtin_amdgcn_global_load_async_to_lds_b128((gv_t*)g, (lv_t*)l, 0, 0);
#else
    *(float4*)l = *(const float4*)g;
#endif
}
static __device__ __forceinline__ void async_wait() {
#if HAVE_ASYNC_LDS
    __builtin_amdgcn_s_wait_asynccnt(0);
#endif
}

// float <-> order-preserving unsigned key (deterministic atomic max/min)
static __device__ __forceinline__ unsigned f2key(float f) {
    unsigned u = __float_as_uint(f);
    return (u & 0x80000000u) ? ~u : (u | 0x80000000u);
}
static __device__ __forceinline__ float key2f(unsigned k) {
    unsigned u = (k & 0x80000000u) ? (k & 0x7FFFFFFFu) : ~k;
    return __uint_as_float(u);
}

// ---------------- squared norms for kNN -----------------------------------
template <int C>
__global__ void k_xx(const float* __restrict__ x, int bC, float* __restrict__ xx) {
    int t = blockIdx.x * blockDim.x + threadIdx.x;
    if (t >= B_ * N_) return;
    int b = t / N_, n = t % N_;
    const float* p = x + (size_t)b * bC * N_ + n;
    float s = 0.f;
#pragma unroll 8
    for (int c = 0; c < C; ++c) { float v = p[(size_t)c * N_]; s += v * v; }
    xx[t] = s;
}

// ---------------- pairwise score GEMM (WMMA f32 16x16x4) ------------------
// scores[b][i][j] = 2*<x_i,x_j> - |x_i|^2 - |x_j|^2   (branch-free inner loop)
template <int C>
__global__ void k_scores(const float* __restrict__ x, int bC,
                         const float* __restrict__ xx, float* __restrict__ sc) {
    constexpr int Cp = (C + 3) & ~3;
    int b = blockIdx.z;
    int i0 = blockIdx.x * 16, j0 = blockIdx.y * 16;
    int lane = threadIdx.x, half = lane >> 4, lr = lane & 15;
    const float* xb = x + (size_t)b * bC * N_;
    int ri = i0 + lr, cj = j0 + lr;
    v8f acc = {};
#pragma unroll 8
    for (int c0 = 0; c0 < Cp; c0 += 4) {
        int ca = c0 + (half << 1);
        float ax, ay, bx, by;
        if (C % 4 == 0) {                       // compile-time: no guards
            ax = xb[(size_t)ca * N_ + ri];
            ay = xb[(size_t)(ca + 1) * N_ + ri];
            bx = xb[(size_t)ca * N_ + cj];
            by = xb[(size_t)(ca + 1) * N_ + cj];
        } else {                                // clamp + select, no exec branches
            int cx = (ca < C) ? ca : 0;
            int cy = (ca + 1 < C) ? ca + 1 : 0;
            float t0 = xb[(size_t)cx * N_ + ri], t1 = xb[(size_t)cy * N_ + ri];
            float t2 = xb[(size_t)cx * N_ + cj], t3 = xb[(size_t)cy * N_ + cj];
            ax = (ca < C) ? t0 : 0.f;  ay = (ca + 1 < C) ? t1 : 0.f;
            bx = (ca < C) ? t2 : 0.f;  by = (ca + 1 < C) ? t3 : 0.f;
        }
        v2f a = {ax, ay}, bm = {bx, by};
        acc = wmma_f32x4(a, bm, acc);
    }
    const float* xxb = xx + (size_t)b * N_;
    float xj = xxb[cj];
#pragma unroll
    for (int r = 0; r < 8; ++r) {
        int i = i0 + r + (half << 3);
        sc[((size_t)b * N_ + i) * N_ + (size_t)(j0 + lr)] = 2.f * acc[r] - xxb[i] - xj;
    }
}

// ---------------- top-k=20 per row (stable, smallest index on ties) -------
__global__ void k_topk(const float* __restrict__ sc, int* __restrict__ idx) {
    __shared__ float row[N_];
    __shared__ float rv[256];
    __shared__ int   ri[256];
    int bn = blockIdx.x;
    int tid = threadIdx.x;
    const float* src = sc + (size_t)bn * N_;
    for (int t = tid; t < N_; t += 256) row[t] = src[t];
    __syncthreads();
    for (int kk = 0; kk < KK_; ++kk) {
        float bv = -3.0e38f; int bi = 0x7FFFFFFF;
        for (int t = tid; t < N_; t += 256) {
            float v = row[t];
            if (v > bv) { bv = v; bi = t; }
        }
        rv[tid] = bv; ri[tid] = bi;
        __syncthreads();
        for (int s = 128; s > 0; s >>= 1) {
            if (tid < s) {
                if (rv[tid + s] > rv[tid] ||
                    (rv[tid + s] == rv[tid] && ri[tid + s] < ri[tid])) {
                    rv[tid] = rv[tid + s]; ri[tid] = ri[tid + s];
                }
            }
            __syncthreads();
        }
        if (tid == 0) {
            idx[(size_t)bn * KK_ + kk] = ri[0];
            row[ri[0]] = -3.0e38f;
        }
        __syncthreads();
    }
}

// ---------------- init max/min key buffers --------------------------------
__global__ void k_init(unsigned* mx, unsigned* mn, int n) {
    int t = blockIdx.x * blockDim.x + threadIdx.x;
    if (t < n) { mx[t] = 0u; mn[t] = 0xFFFFFFFFu; }
}

// ---------------- edge-conv GEMM: y = W(O x 2C) * F(2C x N*K) -------------
template <int C>
__global__ void k_edge_gemm(const float* __restrict__ W, const float* __restrict__ x,
                            int bC, const int* __restrict__ idx,
                            unsigned* mx, unsigned* mn, float* __restrict__ part,
                            int O, int OT, int CB, int Mtiles) {
    constexpr int twoC  = 2 * C;
    constexpr int twoCp = (twoC + 3) & ~3;
    __shared__ float Ft[16][twoCp + 8];        // [col m][channel c]
    __shared__ int   jb[16], nb_[16];
    __shared__ float ssum[16][17], ssq[16][17];
    int b = blockIdx.z, ot = blockIdx.x, cb = blockIdx.y;
    int o0 = ot * 16;
    int lane = threadIdx.x, half = lane >> 4, lr = lane & 15;
    const float* xb = x + (size_t)b * bC * N_;
    float s_sum[8] = {}, s_sq[8] = {};
    const float* Wr = W + (size_t)(o0 + lr) * twoC;

    for (int mt = cb; mt < Mtiles; mt += CB) {
        int m0 = mt * 16;
        if (lane < 16) {
            int m = m0 + lane;
            int n = m / KK_;
            int kk = m - n * KK_;
            jb[lane]  = idx[((size_t)b * N_ + n) * KK_ + kk];
            nb_[lane] = n;
        }
        __syncthreads();
        // neighbor - center differences (VALU path)
        for (int e = lane; e < 16 * C; e += 32) {
            int mm = e / C, c = e - mm * C;
            Ft[mm][c] = xb[(size_t)c * N_ + jb[mm]] - xb[(size_t)c * N_ + nb_[mm]];
        }
        // center copy: pure global->LDS, async engine
        for (int e = lane; e < 16 * C; e += 32) {
            int mm = e / C, c = e - mm * C;
            async_b32(&xb[(size_t)c * N_ + nb_[mm]], &Ft[mm][C + c]);
        }
        if (twoCp > twoC) {                     // zero pad (compile-time dead if none)
            constexpr int PAD = (twoCp > twoC) ? (twoCp - twoC) : 1;
            for (int e = lane; e < 16 * PAD; e += 32) {
                int mm = e / PAD;
                Ft[mm][twoC + (e % PAD)] = 0.f;
            }
        }
        async_wait();
        __syncthreads();
        v8f acc = {};
#pragma unroll 8
        for (int c0 = 0; c0 < twoCp; c0 += 4) {
            int ca = c0 + (half << 1);
            if (c0 + 4 < twoCp) __builtin_prefetch(Wr + c0 + 4, 0, 1);
            float ax, ay;
            if (twoC % 4 == 0) {
                ax = Wr[ca];
                ay = Wr[ca + 1];
            } else {
                int cx = (ca < twoC) ? ca : 0;
                int cy = (ca + 1 < twoC) ? ca + 1 : 0;
                float t0 = Wr[cx], t1 = Wr[cy];
                ax = (ca < twoC) ? t0 : 0.f;
                ay = (ca + 1 < twoC) ? t1 : 0.f;
            }
            v2f a = {ax, ay};
            v2f bm = {Ft[lr][ca], Ft[lr][ca + 1]};
            acc = wmma_f32x4(a, bm, acc);
        }
#pragma unroll
        for (int r = 0; r < 8; ++r) {
            int o = o0 + r + (half << 3);
            int n = nb_[lr];
            float v = acc[r];
            unsigned key = f2key(v);
            size_t si = ((size_t)b * O + o) * N_ + n;
            atomicMax(&mx[si], key);
            atomicMin(&mn[si], key);
            s_sum[r] += v;
            s_sq[r]  += v * v;
        }
        __syncthreads();
    }
#pragma unroll
    for (int r = 0; r < 8; ++r) {
        ssum[r + (half << 3)][lr] = s_sum[r];
        ssq [r + (half << 3)][lr] = s_sq[r];
    }
    __syncthreads();
    if (lane < 16) {
        float ps = 0.f, pq = 0.f;
        for (int c = 0; c < 16; ++c) { ps += ssum[lane][c]; pq += ssq[lane][c]; }
        size_t slot = (((size_t)b * OT + ot) * CB + cb) * 32;
        part[slot + lane]      = ps;
        part[slot + 16 + lane] = pq;
    }
}

// ---------------- deterministic BN stats reduction ------------------------
__global__ void k_reduce(const float* __restrict__ part, const float* __restrict__ g,
                         const float* __restrict__ bb, float* __restrict__ chanp,
                         int O, int OT, int CB, float invCnt) {
    int o = blockIdx.x * blockDim.x + threadIdx.x;
    if (o >= O) return;
    int ot = o >> 4, r = o & 15;
    float s = 0.f, q = 0.f;
    for (int b = 0; b < B_; ++b)
        for (int cb = 0; cb < CB; ++cb) {
            size_t slot = (((size_t)b * OT + ot) * CB + cb) * 32;
            s += part[slot + r];
            q += part[slot + 16 + r];
        }
    float m = s * invCnt;
    float v = q * invCnt - m * m;
    if (v < 0.f) v = 0.f;
    float sc = g[o] * rsqrtf(v + EPS_);
    chanp[o * 4 + 0] = sc;
    chanp[o * 4 + 1] = bb[o] - m * sc;
    chanp[o * 4 + 2] = (g[o] >= 0.f) ? 1.f : 0.f;
}

// ---------------- edge-conv finalize: BN + LeakyReLU + max-over-k ---------
__global__ void k_fin_edge(const unsigned* __restrict__ mx, const unsigned* __restrict__ mn,
                           const float* __restrict__ chanp, float* __restrict__ xcat,
                           int choff, int O, int total) {
    int t = blockIdx.x * blockDim.x + threadIdx.x;
    if (t >= total) return;
    int n = t % N_;
    int rest = t / N_;
    int o = rest % O;
    int b = rest / O;
    float raw = (chanp[o * 4 + 2] > 0.5f) ? key2f(mx[t]) : key2f(mn[t]);
    float yn = raw * chanp[o * 4] + chanp[o * 4 + 1];
    float a = (yn > 0.f) ? yn : NEG_SLOPE_ * yn;
    xcat[((size_t)b * 512 + choff + o) * N_ + n] = a;
}

// ---------------- final 1024x512 GEMM (WMMA) with fused stats -------------
// LDS B-tile is channel-major so both src and dst are 16B-contiguous:
// staged entirely with async b128 global->LDS transfers.
__global__ void k_gemm5(const float* __restrict__ W, const float* __restrict__ xcat,
                        unsigned* mx, unsigned* mn, float* __restrict__ part,
                        float* __restrict__ slice, int O, int OT, int CB, int Mtiles) {
    __shared__ float Ft[512][16];              // [channel c][col m]
    __shared__ float ssum[16][17], ssq[16][17];
    int b = blockIdx.z, ot = blockIdx.x, cb = blockIdx.y;
    int o0 = ot * 16;
    int lane = threadIdx.x, half = lane >> 4, lr = lane & 15;
    const float* xb = xcat + (size_t)b * 512 * N_;
    const float* Wr = W + (size_t)(o0 + lr) * 512;
    float s_sum[8] = {}, s_sq[8] = {};

    for (int mt = cb; mt < Mtiles; mt += CB) {
        int m0 = mt * 16;
        for (int e = lane; e < 512 * 4; e += 32) {
            int c = e >> 2, q = e & 3;
            async_b128(&xb[(size_t)c * N_ + m0 + q * 4], &Ft[c][q * 4]);
        }
        async_wait();
        __syncthreads();
        v8f acc = {};
#pragma unroll 8
        for (int c0 = 0; c0 < 512; c0 += 4) {
            int ca = c0 + (half << 1);
            if (c0 + 4 < 512) __builtin_prefetch(Wr + c0 + 4, 0, 1);
            v2f a  = {Wr[ca], Wr[ca + 1]};
            v2f bm = {Ft[ca][lr], Ft[ca + 1][lr]};
            acc = wmma_f32x4(a, bm, acc);
        }
#pragma unroll
        for (int r = 0; r < 8; ++r) {
            int o = o0 + r + (half << 3);
            int n = m0 + lr;
            float v = acc[r];
            unsigned key = f2key(v);
            int nn = n >> 7;
            size_t si = ((size_t)b * O + o) * 32 + nn;
            atomicMax(&mx[si], key);
            atomicMin(&mn[si], key);
            if (n < 128) slice[((size_t)b * O + o) * 128 + n] = v;
            s_sum[r] += v;
            s_sq[r]  += v * v;
        }
        __syncthreads();
    }
#pragma unroll
    for (int r = 0; r < 8; ++r) {
        ssum[r + (half << 3)][lr] = s_sum[r];
        ssq [r + (half << 3)][lr] = s_sq[r];
    }
    __syncthreads();
    if (lane < 16) {
        float ps = 0.f, pq = 0.f;
        for (int c = 0; c < 16; ++c) { ps += ssum[lane][c]; pq += ssq[lane][c]; }
        size_t slot = (((size_t)b * OT + ot) * CB + cb) * 32;
        part[slot + lane]      = ps;
        part[slot + 16 + lane] = pq;
    }
#if __has_builtin(__builtin_amdgcn_s_wait_tensorcnt)
    __builtin_amdgcn_s_wait_tensorcnt(0);
#endif
}

// ---------------- output 1: pf[..., 0, :] ---------------------------------
__global__ void k_fin5_slice(const float* __restrict__ slice, const float* __restrict__ chanp,
                             float* __restrict__ out, int total) {
    int t = blockIdx.x * blockDim.x + threadIdx.x;
    if (t >= total) return;
    int rest = t / 128;
    int e = rest % 1024;
    float yn = slice[t] * chanp[e * 4] + chanp[e * 4 + 1];
    out[t] = (yn > 0.f) ? yn : NEG_SLOPE_ * yn;
}

// ---------------- output 2: group_features = concat([xm, xm]) -------------
__global__ void k_fin5_gf(const unsigned* __restrict__ mx, const unsigned* __restrict__ mn,
                          const float* __restrict__ chanp, float* __restrict__ out, int total) {
    int t = blockIdx.x * blockDim.x + threadIdx.x;
    if (t >= total) return;
    int nn = t % 32;
    int rest = t / 32;
    int e = rest % 1024;
    int b = rest / 1024;
    float raw = (chanp[e * 4 + 2] > 0.5f) ? key2f(mx[t]) : key2f(mn[t]);
    float yn = raw * chanp[e * 4] + chanp[e * 4 + 1];
    float a = (yn > 0.f) ? yn : NEG_SLOPE_ * yn;
    size_t o1 = ((size_t)b * 2048 + e) * 32 + nn;
    size_t o2 = ((size_t)b * 2048 + 1024 + e) * 32 + nn;
    out[o1] = a;
    out[o2] = a;
}

// ---------------- host-side per-layer driver ------------------------------
template <int C>
static void run_edge_layer(const float* xin, int bC, int O, int choffOut,
                           const float* W, const float* g, const float* bb,
                           float* xx, float* scores, int* idxb,
                           unsigned* mx, unsigned* mn, float* part, float* chanp,
                           float* xcat, hipStream_t stream) {
    const int CBe = 64;
    const int Mte = N_ * KK_ / 16;
    int OT = O / 16;
    int cnt = B_ * O * N_;
    k_xx<C><<<(B_ * N_ + 255) / 256, 256, 0, stream>>>(xin, bC, xx);
    k_scores<C><<<dim3(N_ / 16, N_ / 16, B_), 32, 0, stream>>>(xin, bC, xx, scores);
    k_topk<<<B_ * N_, 256, 0, stream>>>(scores, idxb);
    k_init<<<(cnt + 255) / 256, 256, 0, stream>>>(mx, mn, cnt);
    k_edge_gemm<C><<<dim3(OT, CBe, B_), 32, 0, stream>>>(W, xin, bC, idxb, mx, mn, part,
                                                         O, OT, CBe, Mte);
    k_reduce<<<(O + 255) / 256, 256, 0, stream>>>(part, g, bb, chanp, O, OT, CBe,
                                                  1.f / ((float)B_ * N_ * KK_));
    k_fin_edge<<<(cnt + 255) / 256, 256, 0, stream>>>(mx, mn, chanp, xcat, choffOut, O, cnt);
}

extern "C" void kernel_launch(void* const* d_in, const int* in_sizes, int n_in,
                              void* d_out, int out_size, void* d_ws, size_t ws_size,
                              hipStream_t stream) {
    const float* x0 = (const float*)d_in[0];
    const float* Wl[5] = {(const float*)d_in[1], (const float*)d_in[4], (const float*)d_in[7],
                          (const float*)d_in[10], (const float*)d_in[13]};
    const float* gl[5] = {(const float*)d_in[2], (const float*)d_in[5], (const float*)d_in[8],
                          (const float*)d_in[11], (const float*)d_in[14]};
    const float* bl[5] = {(const float*)d_in[3], (const float*)d_in[6], (const float*)d_in[9],
                          (const float*)d_in[12], (const float*)d_in[15]};

    float* ws = (float*)d_ws;
    size_t offScores = 0;
    size_t offXX   = offScores + (size_t)B_ * N_ * N_;
    size_t offIdx  = offXX   + (size_t)B_ * N_;
    size_t offXcat = offIdx  + (size_t)B_ * N_ * KK_;
    size_t offMax  = offXcat + (size_t)B_ * 512 * N_;
    size_t offMin  = offMax  + (size_t)B_ * 256 * N_;
    size_t offPart = offMin  + (size_t)B_ * 256 * N_;
    size_t offChan = offPart + 262144;
    size_t offSlice= offChan + 4096;

    float*    scores = ws + offScores;
    float*    xx     = ws + offXX;
    int*      idxb   = (int*)(ws + offIdx);
    float*    xcat   = ws + offXcat;
    unsigned* mx     = (unsigned*)(ws + offMax);
    unsigned* mn     = (unsigned*)(ws + offMin);
    float*    part   = ws + offPart;
    float*    chanp  = ws + offChan;
    float*    slice  = ws + offSlice;
    float*    out    = (float*)d_out;

    run_edge_layer<3>  (x0,                    3,   64,  0,   Wl[0], gl[0], bl[0],
                        xx, scores, idxb, mx, mn, part, chanp, xcat, stream);
    run_edge_layer<64> (xcat,                  512, 64,  64,  Wl[1], gl[1], bl[1],
                        xx, scores, idxb, mx, mn, part, chanp, xcat, stream);
    run_edge_layer<64> (xcat + (size_t)64*N_,  512, 128, 128, Wl[2], gl[2], bl[2],
                        xx, scores, idxb, mx, mn, part, chanp, xcat, stream);
    run_edge_layer<128>(xcat + (size_t)128*N_, 512, 256, 256, Wl[3], gl[3], bl[3],
                        xx, scores, idxb, mx, mn, part, chanp, xcat, stream);

    // layer 5: pointwise 1024x512 over xcat
    const int O5 = 1024, OT5 = 64, CB5 = 32, Mt5 = N_ / 16;
    int cnt5 = B_ * O5 * 32;
    k_init<<<(cnt5 + 255) / 256, 256, 0, stream>>>(mx, mn, cnt5);
    k_gemm5<<<dim3(OT5, CB5, B_), 32, 0, stream>>>(Wl[4], xcat, mx, mn, part, slice,
                                                   O5, OT5, CB5, Mt5);
    k_reduce<<<(O5 + 255) / 256, 256, 0, stream>>>(part, gl[4], bl[4], chanp, O5, OT5, CB5,
                                                   1.f / ((float)B_ * N_));
    int nslice = B_ * O5 * 128;
    k_fin5_slice<<<(nslice + 255) / 256, 256, 0, stream>>>(slice, chanp, out, nslice);
    k_fin5_gf<<<(cnt5 + 255) / 256, 256, 0, stream>>>(mx, mn, chanp, out + (size_t)nslice, cnt5);
}